// TransD_38517266710783
// MI455X (gfx1250) — compile-verified
//
#include <hip/hip_runtime.h>

// MI455X / gfx1250, wave32.
// TransD scoring, B=500k items, ENT_DIM=REL_DIM=128, 500 relations.
//
// Strategy:
//  1) Precompute per-relation scalars  P2=sum(rp^2), R2=sum(r^2), PR=sum(rp*r),
//     SP=sum(rp), SR=sum(r)  into d_ws (500 x 8 floats = 16 KB).
//  2) Main kernel: 16 items per wave. Per item need only
//        c = dot(h_p,h) - dot(t_p,t),  s = sum(h) - sum(t)
//     score = c^2*P2 + R2 + 128*s^2 + 2c*PR + 2cs*SP + 2s*SR   (exact expansion).
//     The two Gram matrices are accumulated with V_WMMA_F32_16X16X4_F32 over
//     32 K-chunks; diagonals extracted at the end. Lane L loads dims
//     (4k + 2*(L>>4), +1) of item (L&15): this IS the ISA A/B layout for
//     f32 16x16x4, one global_load_b64 per matrix per chunk.

typedef __attribute__((ext_vector_type(2))) float v2f;
typedef __attribute__((ext_vector_type(4))) float v4f;
typedef __attribute__((ext_vector_type(8))) float v8f;

#define ENT_DIM 128
#define REL_PAD 8   // floats per relation record in ws

// ---------------- Kernel 1: relation scalar precompute (one wave per relation)
__global__ void transd_rel_precompute(const float* __restrict__ rel_emb,
                                      const float* __restrict__ rel_emb_p,
                                      float* __restrict__ ws, int n_rels) {
    int wave = (int)((blockIdx.x * blockDim.x + threadIdx.x) >> 5);
    int lane = (int)(threadIdx.x & 31);
    if (wave >= n_rels) return;

    const v4f r4  = *(const v4f*)(rel_emb   + (size_t)wave * ENT_DIM + lane * 4);
    const v4f rp4 = *(const v4f*)(rel_emb_p + (size_t)wave * ENT_DIM + lane * 4);

    float p2 = rp4.x*rp4.x + rp4.y*rp4.y + rp4.z*rp4.z + rp4.w*rp4.w;
    float r2 = r4.x*r4.x  + r4.y*r4.y  + r4.z*r4.z  + r4.w*r4.w;
    float pr = rp4.x*r4.x + rp4.y*r4.y + rp4.z*r4.z + rp4.w*r4.w;
    float sp = rp4.x + rp4.y + rp4.z + rp4.w;
    float sr = r4.x  + r4.y  + r4.z  + r4.w;

#pragma unroll
    for (int off = 16; off > 0; off >>= 1) {
        p2 += __shfl_xor(p2, off, 32);
        r2 += __shfl_xor(r2, off, 32);
        pr += __shfl_xor(pr, off, 32);
        sp += __shfl_xor(sp, off, 32);
        sr += __shfl_xor(sr, off, 32);
    }
    if (lane == 0) {
        float* q = ws + (size_t)wave * REL_PAD;
        q[0] = p2; q[1] = r2; q[2] = pr; q[3] = sp; q[4] = sr;
    }
}

// ---------------- Kernel 2: 16 items per wave, WMMA Gram accumulation
__global__ void transd_score_wmma(const float* __restrict__ head_emb,
                                  const float* __restrict__ head_emb_p,
                                  const float* __restrict__ tail_emb,
                                  const float* __restrict__ tail_emb_p,
                                  const int* __restrict__ head_indices,
                                  const int* __restrict__ tail_indices,
                                  const int* __restrict__ rel_indices,
                                  const float* __restrict__ rel_ws,
                                  float* __restrict__ out, int Bsz) {
    int waveg = (int)((blockIdx.x * blockDim.x + threadIdx.x) >> 5);
    int lane  = (int)(threadIdx.x & 31);
    int nTiles = (Bsz + 15) >> 4;
    if (waveg >= nTiles) return;          // wave-uniform exit

    const int tile16 = waveg << 4;
    int item = tile16 + (lane & 15);
    if (item >= Bsz) item = Bsz - 1;      // clamp (data-level, EXEC stays full)
    const int half2 = (lane >> 4) << 1;   // 0 for lanes 0-15, 2 for lanes 16-31

    const long hrow = (long)head_indices[item] * ENT_DIM;
    const long trow = (long)tail_indices[item] * ENT_DIM;
    const float* ph  = head_emb   + hrow + half2;
    const float* php = head_emb_p + hrow + half2;
    const float* pt  = tail_emb   + trow + half2;
    const float* ptp = tail_emb_p + trow + half2;

    v8f accH = {0.f,0.f,0.f,0.f,0.f,0.f,0.f,0.f};
    v8f accT = {0.f,0.f,0.f,0.f,0.f,0.f,0.f,0.f};
    float sh = 0.f, st = 0.f;

#pragma unroll
    for (int k = 0; k < 32; ++k) {
        // Lane L holds dims (4k+half2, 4k+half2+1) of item (L&15):
        // exactly the f32 16x16x4 A (MxK) and B (KxN) VGPR layouts.
        v2f hp2 = *(const v2f*)(php + 4 * k);
        v2f h2  = *(const v2f*)(ph  + 4 * k);
        v2f tp2 = *(const v2f*)(ptp + 4 * k);
        v2f t2  = *(const v2f*)(pt  + 4 * k);

        accH = __builtin_amdgcn_wmma_f32_16x16x4_f32(
                   false, hp2, false, h2, (short)0, accH, false, false);
        accT = __builtin_amdgcn_wmma_f32_16x16x4_f32(
                   false, tp2, false, t2, (short)0, accT, false, false);

        sh += h2.x + h2.y;   // running partial of sum(h[item])
        st += t2.x + t2.y;
    }

    // Row sums: halves of item (L&15) live in lanes L and L^16.
    sh += __shfl_xor(sh, 16, 32);
    st += __shfl_xor(st, 16, 32);
    const float s = sh - st;

    // Diagonal extraction: C vgpr j holds (M=j, N=lane) lanes 0-15 and
    // (M=j+8, N=lane-16) lanes 16-31  ->  C[j,j] @ (vgpr j, lane j),
    // C[j+8,j+8] @ (vgpr j, lane j+24).
    float c = 0.f;
    const int l16 = lane & 15;
#pragma unroll
    for (int j = 0; j < 8; ++j) {
        float dH = accH[j];
        float dT = accT[j];
        float lo = __shfl(dH, j, 32)      - __shfl(dT, j, 32);
        float hi = __shfl(dH, j + 24, 32) - __shfl(dT, j + 24, 32);
        c = (l16 == j)     ? lo : c;
        c = (l16 == j + 8) ? hi : c;
    }

    // Lanes 0..15 emit their item's score (all WMMAs are done; divergence OK).
    const int oi = tile16 + lane;
    if (lane < 16 && oi < Bsz) {
        const float* q = rel_ws + (size_t)rel_indices[oi] * REL_PAD;
        const float P2 = q[0], R2 = q[1], PR = q[2], SP = q[3], SR = q[4];
        out[oi] = c * c * P2 + R2 + 128.f * s * s
                + 2.f * c * PR + 2.f * c * s * SP + 2.f * s * SR;
    }
}

extern "C" void kernel_launch(void* const* d_in, const int* in_sizes, int n_in,
                              void* d_out, int out_size, void* d_ws, size_t ws_size,
                              hipStream_t stream) {
    const float* head_emb   = (const float*)d_in[0];
    const float* head_emb_p = (const float*)d_in[1];
    const float* tail_emb   = (const float*)d_in[2];
    const float* tail_emb_p = (const float*)d_in[3];
    const float* rel_emb    = (const float*)d_in[4];
    const float* rel_emb_p  = (const float*)d_in[5];
    const int*   head_idx   = (const int*)d_in[6];
    const int*   tail_idx   = (const int*)d_in[7];
    const int*   rel_idx    = (const int*)d_in[8];
    float*       out        = (float*)d_out;
    float*       rel_ws     = (float*)d_ws;   // needs n_rels*8*4 = 16 KB

    const int n_rels = in_sizes[4] / ENT_DIM;  // 500
    const int Bsz    = in_sizes[6];            // 500000

    // Kernel 1: one wave (32 threads) per relation.
    {
        const int threads = 256;                       // 8 waves/block
        const int blocks  = (n_rels * 32 + threads - 1) / threads;
        transd_rel_precompute<<<blocks, threads, 0, stream>>>(
            rel_emb, rel_emb_p, rel_ws, n_rels);
    }

    // Kernel 2: one wave per 16 items.
    {
        const int nTiles  = (Bsz + 15) >> 4;           // 31250
        const int threads = 256;                       // 8 waves/block
        const int blocks  = (nTiles + 7) / 8;
        transd_score_wmma<<<blocks, threads, 0, stream>>>(
            head_emb, head_emb_p, tail_emb, tail_emb_p,
            head_idx, tail_idx, rel_idx, rel_ws, out, Bsz);
    }
}